// CusGATConv_38397007626854
// MI455X (gfx1250) — compile-verified
//
#include <hip/hip_runtime.h>
#include <hip/hip_bf16.h>

typedef __attribute__((ext_vector_type(2))) float v2f;
typedef __attribute__((ext_vector_type(8))) float v8f;

#define WMMA_F32K4(a, b, c) \
  __builtin_amdgcn_wmma_f32_16x16x4_f32(false, (a), false, (b), (short)0, (c), false, false)

static constexpr int HEADS = 4;
static constexpr int C     = 64;   // C_IN == C_OUT
static constexpr int EIW   = 32;   // E_IN
static constexpr int EOW   = 32;   // E_OUT

// ---------------------------------------------------------------------------
// sign-aware fp32 atomic max (memory initialized to -inf)
// ---------------------------------------------------------------------------
__device__ __forceinline__ void atomicMaxF(float* addr, float v) {
  if (v >= 0.0f) atomicMax((int*)addr, __float_as_int(v));
  else           atomicMin((unsigned int*)addr, (unsigned int)__float_as_int(v));
}

// ---------------------------------------------------------------------------
// cooperative stage of B[K,Nc] (row-major, ldb) into LDS, column-major:
//   Bs[col*K + k] = B[k*ldb + col]
// global reads coalesced across threads; one-time LDS write.
// ---------------------------------------------------------------------------
__device__ __forceinline__ void stageB(const float* __restrict__ B, int ldb,
                                       float* __restrict__ Bs, int K, int Nc) {
  for (int idx = (int)threadIdx.x; idx < K * Nc; idx += (int)blockDim.x) {
    int k = idx / Nc, col = idx - k * Nc;
    Bs[col * K + k] = B[k * ldb + col];
  }
  __syncthreads();
}

// ---------------------------------------------------------------------------
// generic fp32 WMMA GEMM: C[M,Nc](ldc) = A[M,K](lda) @ B[K,Nc](ldb) + bias
// one 16x16 output tile per wave32; weights staged in LDS (ds_load_b64 path).
// M % 16 == 0, K % 4 == 0, Nc % 16 == 0, K*Nc <= 4096.
// ---------------------------------------------------------------------------
__global__ void k_gemm_wmma(const float* __restrict__ A, int lda,
                            const float* __restrict__ B, int ldb,
                            const float* __restrict__ bias,
                            float* __restrict__ Cmat, int ldc,
                            int M, int K, int Nc, int accumulate) {
  __shared__ float Bs[64 * 64];  // 16 KB max (K<=64, Nc<=64)
  stageB(B, ldb, Bs, K, Nc);

  int wave = (int)((blockIdx.x * blockDim.x + threadIdx.x) >> 5);
  int lane = (int)(threadIdx.x & 31);
  int ntn  = Nc >> 4;
  int tiles = (M >> 4) * ntn;
  if (wave >= tiles) return;
  int tm = wave / ntn, tn = wave - tm * ntn;

  int arow  = tm * 16 + (lane & 15);   // A row for this lane
  int bcol  = tn * 16 + (lane & 15);   // B column for this lane
  int khalf = (lane >> 4) * 2;         // K sub-offset per lane half

  // speculative prefetch of the next A tile (same lane pattern)
  if ((tm + 1) * 16 < M) __builtin_prefetch(&A[(arow + 16) * lda], 0, 1);

  const float* __restrict__ bsc = &Bs[bcol * K];
  v8f acc = {};
  for (int k0 = 0; k0 < K; k0 += 4) {
    v2f a, b;
    a.x = A[arow * lda + k0 + khalf];
    a.y = A[arow * lda + k0 + khalf + 1];
    b.x = bsc[k0 + khalf];
    b.y = bsc[k0 + khalf + 1];
    acc = WMMA_F32K4(a, b, acc);
  }

  int n  = tn * 16 + (lane & 15);
  float bv = bias ? bias[n] : 0.0f;
  for (int g = 0; g < 8; ++g) {
    int m = tm * 16 + g + 8 * (lane >> 4);
    float v = acc[g] + bv;
    if (accumulate) Cmat[m * ldc + n] += v;
    else            Cmat[m * ldc + n]  = v;
  }
}

// ---------------------------------------------------------------------------
// fold edge-output weights:
//   GA[h] = Weg_h[0:64]  @ Wed_h   (64x32)   (gathers on row/source)
//   GB[h] = Weg_h[64:128]@ Wed_h   (64x32)   (gathers on col/target)
//   M1    = Wed[0:32] + sum_h Weg_h[128:160] @ Wed_h   (32x32)
//   cvec  = bed + sum_h beg_h @ Wed_h                  (32)
// ---------------------------------------------------------------------------
__global__ void k_fold_weights(const float* __restrict__ Weg,  // [H,160,32]
                               const float* __restrict__ beg,  // [H,32]
                               const float* __restrict__ Wed,  // [160,32]
                               const float* __restrict__ bed,  // [32]
                               float* __restrict__ GA, float* __restrict__ GB,
                               float* __restrict__ M1, float* __restrict__ cvec) {
  int t = (int)threadIdx.x;  // single 256-thread block
  for (int idx = t; idx < HEADS * 64 * EOW; idx += 256) {
    int h = idx >> 11, r = (idx >> 5) & 63, j = idx & 31;
    const float* WegH = Weg + h * 160 * EOW;
    const float* WedH = Wed + (EIW + EOW * h) * EOW;
    float sa = 0.f, sb = 0.f;
    for (int k = 0; k < EOW; ++k) {
      float w = WedH[k * EOW + j];
      sa += WegH[r * EOW + k] * w;
      sb += WegH[(64 + r) * EOW + k] * w;
    }
    GA[idx] = sa; GB[idx] = sb;
  }
  for (int idx = t; idx < EIW * EOW; idx += 256) {
    int i = idx >> 5, j = idx & 31;
    float accv = Wed[i * EOW + j];
    for (int h = 0; h < HEADS; ++h) {
      const float* WegH = Weg + h * 160 * EOW;
      const float* WedH = Wed + (EIW + EOW * h) * EOW;
      for (int k = 0; k < EOW; ++k)
        accv += WegH[(128 + i) * EOW + k] * WedH[k * EOW + j];
    }
    M1[idx] = accv;
  }
  for (int j = t; j < EOW; j += 256) {
    float accv = bed[j];
    for (int h = 0; h < HEADS; ++h) {
      const float* WedH = Wed + (EIW + EOW * h) * EOW;
      for (int k = 0; k < EOW; ++k) accv += beg[h * EOW + k] * WedH[k * EOW + j];
    }
    cvec[j] = accv;
  }
}

__global__ void k_fill(float* __restrict__ p, int n, float v) {
  int i = (int)(blockIdx.x * blockDim.x + threadIdx.x);
  if (i < n) p[i] = v;
}

// ---------------------------------------------------------------------------
// edge attention pass (per head), one 16(edge)x16(chan) tile per wave.
// alpha_pre = leaky_relu( ea@WatC + bat + Ai[col] + Aj[row], 0.2 )
//   mode 0: atomicMax of alpha_pre into m[col]
//   mode 1: e = exp(alpha_pre - m[col]); s[col]+=e; u[col]+=e*xf[row]
// ---------------------------------------------------------------------------
__global__ void k_edge_alpha(const float* __restrict__ EA,    // [E,32]
                             const float* __restrict__ WatC,  // [32 x 64], ld 64
                             const float* __restrict__ bat,   // [64]
                             const float* __restrict__ Ai,    // [N,64] gather by col
                             const float* __restrict__ Aj,    // [N,64] gather by row
                             const int*   __restrict__ rowi,  // edge_index[0]
                             const int*   __restrict__ coli,  // edge_index[1]
                             const float* __restrict__ xf,    // [N,64]
                             float* __restrict__ m,
                             float* __restrict__ s,
                             float* __restrict__ u,
                             int E, int mode) {
  __shared__ float Bs[EIW * C];  // 8 KB, column-major
  stageB(WatC, C, Bs, EIW, C);

  int wave = (int)((blockIdx.x * blockDim.x + threadIdx.x) >> 5);
  int lane = (int)(threadIdx.x & 31);
  const int ntn = C / 16;  // 4
  int tiles = (E >> 4) * ntn;
  if (wave >= tiles) return;
  int tm = wave / ntn, tn = wave - tm * ntn;

  int erow  = tm * 16 + (lane & 15);
  int bcol  = tn * 16 + (lane & 15);
  int khalf = (lane >> 4) * 2;

  if ((tm + 1) * 16 < E) __builtin_prefetch(&EA[(erow + 16) * EIW], 0, 1);

  const float* __restrict__ bsc = &Bs[bcol * EIW];
  v8f acc = {};
  for (int k0 = 0; k0 < EIW; k0 += 4) {
    v2f a, b;
    a.x = EA[erow * EIW + k0 + khalf];
    a.y = EA[erow * EIW + k0 + khalf + 1];
    b.x = bsc[k0 + khalf];
    b.y = bsc[k0 + khalf + 1];
    acc = WMMA_F32K4(a, b, acc);
  }

  int ch = tn * 16 + (lane & 15);
  float bv = bat[ch];
  for (int g = 0; g < 8; ++g) {
    int e = tm * 16 + g + 8 * (lane >> 4);
    int r = rowi[e], c = coli[e];
    float val = acc[g] + bv + Ai[c * C + ch] + Aj[r * C + ch];
    val = (val >= 0.0f) ? val : 0.2f * val;  // leaky_relu(0.2)
    if (mode == 0) {
      atomicMaxF(&m[c * C + ch], val);
    } else {
      float mm = m[c * C + ch];
      if (mm < -1e30f) mm = 0.0f;            // empty-segment clip
      float ev = __expf(val - mm);
      atomicAdd(&s[c * C + ch], ev);
      atomicAdd(&u[c * C + ch], ev * xf[r * C + ch]);
    }
  }
}

__global__ void k_upd_div(const float* __restrict__ u, const float* __restrict__ s,
                          float* __restrict__ upd, int n) {
  int i = (int)(blockIdx.x * blockDim.x + threadIdx.x);
  if (i < n) upd[i] = u[i] / (s[i] + 1e-16f);
}

// ---------------------------------------------------------------------------
// out_edge[e] = ea[e]@M1 + P[row[e]] + Q[col[e]] + cvec
// ---------------------------------------------------------------------------
__global__ void k_edge_out(const float* __restrict__ EA,
                           const float* __restrict__ M1,
                           const float* __restrict__ cvec,
                           const float* __restrict__ P,
                           const float* __restrict__ Q,
                           const int*   __restrict__ rowi,
                           const int*   __restrict__ coli,
                           float* __restrict__ out_edge, int E) {
  __shared__ float Bs[EIW * EOW];  // 4 KB, column-major
  stageB(M1, EOW, Bs, EIW, EOW);

  int wave = (int)((blockIdx.x * blockDim.x + threadIdx.x) >> 5);
  int lane = (int)(threadIdx.x & 31);
  const int ntn = EOW / 16;  // 2
  int tiles = (E >> 4) * ntn;
  if (wave >= tiles) return;
  int tm = wave / ntn, tn = wave - tm * ntn;

  int erow  = tm * 16 + (lane & 15);
  int bcol  = tn * 16 + (lane & 15);
  int khalf = (lane >> 4) * 2;

  if ((tm + 1) * 16 < E) __builtin_prefetch(&EA[(erow + 16) * EIW], 0, 1);

  const float* __restrict__ bsc = &Bs[bcol * EIW];
  v8f acc = {};
  for (int k0 = 0; k0 < EIW; k0 += 4) {
    v2f a, b;
    a.x = EA[erow * EIW + k0 + khalf];
    a.y = EA[erow * EIW + k0 + khalf + 1];
    b.x = bsc[k0 + khalf];
    b.y = bsc[k0 + khalf + 1];
    acc = WMMA_F32K4(a, b, acc);
  }

  int ch = tn * 16 + (lane & 15);
  float cv = cvec[ch];
  for (int g = 0; g < 8; ++g) {
    int e = tm * 16 + g + 8 * (lane >> 4);
    int r = rowi[e], c = coli[e];
    out_edge[e * EOW + ch] = acc[g] + cv + P[r * EOW + ch] + Q[c * EOW + ch];
  }
}

// ---------------------------------------------------------------------------
// host-side orchestration
// ---------------------------------------------------------------------------
extern "C" void kernel_launch(void* const* d_in, const int* in_sizes, int n_in,
                              void* d_out, int out_size, void* d_ws, size_t ws_size,
                              hipStream_t stream) {
  const float* x    = (const float*)d_in[0];
  const int*   ei   = (const int*)  d_in[1];
  const float* ea   = (const float*)d_in[2];
  const float* Wnb  = (const float*)d_in[3];
  const float* bnb  = (const float*)d_in[4];
  const float* Wpi  = (const float*)d_in[5];
  const float* bpi  = (const float*)d_in[6];
  const float* Wpj  = (const float*)d_in[7];
  const float* bpj  = (const float*)d_in[8];
  const float* Weg  = (const float*)d_in[9];
  const float* beg  = (const float*)d_in[10];
  const float* Wat  = (const float*)d_in[11];
  const float* bat  = (const float*)d_in[12];
  const float* Wpt  = (const float*)d_in[13];
  const float* bpt  = (const float*)d_in[14];
  const float* Wed  = (const float*)d_in[15];
  const float* bed  = (const float*)d_in[16];

  const int N = in_sizes[0] / C;       // 50000
  const int E = in_sizes[2] / EIW;     // 400000
  const int N64 = N * C;
  const int* rowi = ei;                // source
  const int* coli = ei + E;            // target

  // workspace carve-up (floats)
  float* ws   = (float*)d_ws;
  float* xf   = ws;                 // [N,64]
  float* tmp  = xf   + N64;         // [N,64]
  float* Ai   = tmp  + N64;         // [N,64]
  float* Aj   = Ai   + N64;         // [N,64]
  float* mbuf = Aj   + N64;         // [N,64]
  float* sbuf = mbuf + N64;         // [N,64]
  float* ubuf = sbuf + N64;         // [N,64]
  float* upd  = ubuf + N64;         // [H,N,64]
  float* P    = upd  + HEADS * N64; // [N,32]
  float* Q    = P    + N * EOW;     // [N,32]
  float* GA   = Q    + N * EOW;     // [H,64,32]
  float* GB   = GA   + HEADS * 64 * EOW;
  float* M1   = GB   + HEADS * 64 * EOW; // [32,32]
  float* cvec = M1   + EOW * EOW;        // [32]

  float* out_pt   = (float*)d_out;       // [N,64]
  float* out_edge = out_pt + N64;        // [E,32]

  auto gemm = [&](const float* A, int lda, const float* B, int ldb,
                  const float* bias, float* Cm, int ldc,
                  int M, int K, int Nc, int accum) {
    int tiles  = (M / 16) * (Nc / 16);
    int blocks = (tiles * 32 + 255) / 256;
    k_gemm_wmma<<<blocks, 256, 0, stream>>>(A, lda, B, ldb, bias, Cm, ldc, M, K, Nc, accum);
  };
  auto fill = [&](float* p, int n, float v) {
    k_fill<<<(n + 255) / 256, 256, 0, stream>>>(p, n, v);
  };

  k_fold_weights<<<1, 256, 0, stream>>>(Weg, beg, Wed, bed, GA, GB, M1, cvec);

  const int edgeBlocksA = ((E / 16) * (C / 16) * 32 + 255) / 256;
  const float NEG_INF = -__builtin_inff();

  for (int h = 0; h < HEADS; ++h) {
    const float* WnbH  = Wnb + h * C * C;
    const float* WpiH  = Wpi + h * C * C;
    const float* WpjH  = Wpj + h * C * C;
    const float* WatH  = Wat + h * 160 * C;     // [160,64]
    const float* WatA  = WatH;                  // rows 0:64   (pi, gather col)
    const float* WatB  = WatH + 64 * C;         // rows 64:128 (pj, gather row)
    const float* WatCc = WatH + 128 * C;        // rows 128:160 (edge_attr)

    // node-level projections (all WMMA f32)
    gemm(x,   C, WnbH, C, bnb + h * C, xf,  C, N, C, C, 0);   // xf = x@Wnb+b
    gemm(xf,  C, WpiH, C, bpi + h * C, tmp, C, N, C, C, 0);   // qi
    gemm(tmp, C, WatA, C, nullptr,     Ai,  C, N, C, C, 0);   // Ai = qi@WatA
    gemm(xf,  C, WpjH, C, bpj + h * C, tmp, C, N, C, C, 0);   // qj
    gemm(tmp, C, WatB, C, nullptr,     Aj,  C, N, C, C, 0);   // Aj = qj@WatB

    fill(mbuf, N64, NEG_INF);
    fill(sbuf, N64, 0.0f);
    fill(ubuf, N64, 0.0f);

    // pass 1: segment max; pass 2: exp + segment sums (recompute, no alpha store)
    k_edge_alpha<<<edgeBlocksA, 256, 0, stream>>>(ea, WatCc, bat + h * C, Ai, Aj,
                                                  rowi, coli, xf, mbuf, sbuf, ubuf, E, 0);
    k_edge_alpha<<<edgeBlocksA, 256, 0, stream>>>(ea, WatCc, bat + h * C, Ai, Aj,
                                                  rowi, coli, xf, mbuf, sbuf, ubuf, E, 1);

    k_upd_div<<<(N64 + 255) / 256, 256, 0, stream>>>(ubuf, sbuf, upd + h * N64, N64);
  }

  // out_pt = x @ Wpt[0:64] + sum_h upd_h @ Wpt[64+64h : 128+64h] + bpt
  gemm(x, C, Wpt, C, bpt, out_pt, C, N, C, C, 0);
  for (int h = 0; h < HEADS; ++h)
    gemm(upd + h * N64, C, Wpt + (C + C * h) * C, C, nullptr, out_pt, C, N, C, C, 1);

  // P = sum_h upd_h @ GA_h ; Q = sum_h upd_h @ GB_h
  fill(P, N * EOW, 0.0f);
  fill(Q, N * EOW, 0.0f);
  for (int h = 0; h < HEADS; ++h) {
    gemm(upd + h * N64, C, GA + h * 64 * EOW, EOW, nullptr, P, EOW, N, C, EOW, 1);
    gemm(upd + h * N64, C, GB + h * 64 * EOW, EOW, nullptr, Q, EOW, N, C, EOW, 1);
  }

  // out_edge = ea @ M1 + P[row] + Q[col] + cvec
  const int edgeBlocksO = ((E / 16) * (EOW / 16) * 32 + 255) / 256;
  k_edge_out<<<edgeBlocksO, 256, 0, stream>>>(ea, M1, cvec, P, Q, rowi, coli, out_edge, E);
}